// Seq2Seq_30932354466279
// MI455X (gfx1250) — compile-verified
//
#include <hip/hip_runtime.h>
#include <math.h>

#define H     1024
#define DV    4096
#define DW    512
#define G4    4096      // 4*H gate rows
#define VOCAB 32000
#define STEPS 39        // MAX_LEN - 1 decode steps
#define MPAD  48        // 39 padded up to 3 WMMA M-tiles

#define NWG   32        // workgroups cooperating on the recurrence
#define TPB   512       // threads per recurrence workgroup (16 waves)
#define EPW   (H / NWG) // 32 hidden elements owned per workgroup
#define LROWS (4 * EPW) // 128 gate rows computed per workgroup

typedef __attribute__((ext_vector_type(2))) float v2f;
typedef __attribute__((ext_vector_type(8))) float v8f;

__device__ __forceinline__ float sigm(float x) { return 1.0f / (1.0f + expf(-x)); }

// Device-wide barrier: atomic arrive + L2 spin with s_sleep backoff.
// `target` must increase by NWG on every use.
__device__ __forceinline__ void grid_barrier(unsigned* cnt, unsigned target)
{
    __syncthreads();
    if (threadIdx.x == 0) {
        __threadfence();   // release prior global stores to device scope
        __hip_atomic_fetch_add(cnt, 1u, __ATOMIC_ACQ_REL, __HIP_MEMORY_SCOPE_AGENT);
        while (__hip_atomic_load(cnt, __ATOMIC_ACQUIRE, __HIP_MEMORY_SCOPE_AGENT) < target)
            __builtin_amdgcn_s_sleep(2);
    }
    __syncthreads();
    __threadfence();       // acquire: invalidate stale L0 lines in every wave
}

// ---------------------------------------------------------------------------
// Kernel 0: zero the barrier counter and the pad rows of H2.
// ---------------------------------------------------------------------------
__global__ void init_kernel(float* __restrict__ H2, unsigned* __restrict__ cnt)
{
    int t = threadIdx.x;
    if (t == 0) *cnt = 0u;
    for (int i = t; i < (MPAD - STEPS) * H; i += 256) H2[STEPS * H + i] = 0.f;
}

// ---------------------------------------------------------------------------
// Kernel 1: LSTM recurrence across NWG cooperating workgroups.
// WG w owns hidden elements j in [32w, 32w+32): it computes gate rows
// {g*1024 + j} for all four gates, applies the activation locally, and
// publishes its h-slice; 2 device barriers per step.
// ---------------------------------------------------------------------------
__global__ __launch_bounds__(TPB) void lstm_seq_kernel(
    const float* __restrict__ vid,
    const float* __restrict__ w_ih1, const float* __restrict__ w_hh1,
    const float* __restrict__ b_ih1, const float* __restrict__ b_hh1,
    const float* __restrict__ w_ih2, const float* __restrict__ w_hh2,
    const float* __restrict__ b_ih2, const float* __restrict__ b_hh2,
    const float* __restrict__ emb,
    float* __restrict__ H2out,
    float* __restrict__ g_h1, float* __restrict__ g_h2,
    float* __restrict__ g_redv, int* __restrict__ g_redi,
    unsigned* __restrict__ cnt)
{
    __shared__ float s_x[DV];        // vid feats (encode only)
    __shared__ float s_h1[H];
    __shared__ float s_h2[H];
    __shared__ float s_word[DW];
    __shared__ float s_gates[LROWS]; // l = g*32 + e
    __shared__ int   s_idx;

    const int tid  = threadIdx.x;
    const int lane = tid & 31;
    const int wave = tid >> 5;       // 0..15
    const int wg   = blockIdx.x;     // 0..NWG-1
    const int e0   = wg * EPW;       // first owned hidden element

    unsigned bar = 0;                // barrier epoch (counts arrivals)
    float c1 = 0.f, c2 = 0.f;        // cell state, held by wave 0 lanes

    for (int i = tid; i < DV; i += TPB) s_x[i] = vid[i];
    __syncthreads();

    // ---- encode LSTM1: gates = vid @ w_ih1^T + b_ih1 + b_hh1  (h = 0)
    for (int l = wave; l < LROWS; l += 16) {
        const int r = (l >> 5) * H + e0 + (l & 31);
        const float* wr = w_ih1 + (size_t)r * DV;
        float acc = 0.f;
        for (int k = lane * 4; k < DV; k += 128) {
            float4 w4 = *(const float4*)(wr + k);
            acc += w4.x * s_x[k] + w4.y * s_x[k + 1] + w4.z * s_x[k + 2] + w4.w * s_x[k + 3];
        }
        for (int off = 16; off; off >>= 1) acc += __shfl_xor(acc, off, 32);
        if (lane == 0) s_gates[l] = acc + b_ih1[r] + b_hh1[r];
    }
    __syncthreads();
    if (wave == 0) {
        float ig = sigm(s_gates[lane]);
        float fg = sigm(s_gates[EPW + lane]);
        float gg = tanhf(s_gates[2 * EPW + lane]);
        float og = sigm(s_gates[3 * EPW + lane]);
        c1 = fg * c1 + ig * gg;
        g_h1[e0 + lane] = og * tanhf(c1);
    }
    bar += NWG; grid_barrier(cnt, bar);
    for (int i = tid; i < H; i += TPB) s_h1[i] = g_h1[i];
    __syncthreads();

    // ---- encode LSTM2: inp2 = [h1, 0-word], h2 = 0
    for (int l = wave; l < LROWS; l += 16) {
        const int r = (l >> 5) * H + e0 + (l & 31);
        const float* wr = w_ih2 + (size_t)r * (H + DW);
        float acc = 0.f;
        for (int k = lane * 4; k < H; k += 128) {
            float4 w4 = *(const float4*)(wr + k);
            acc += w4.x * s_h1[k] + w4.y * s_h1[k + 1] + w4.z * s_h1[k + 2] + w4.w * s_h1[k + 3];
        }
        for (int off = 16; off; off >>= 1) acc += __shfl_xor(acc, off, 32);
        if (lane == 0) s_gates[l] = acc + b_ih2[r] + b_hh2[r];
    }
    __syncthreads();
    if (wave == 0) {
        float ig = sigm(s_gates[lane]);
        float fg = sigm(s_gates[EPW + lane]);
        float gg = tanhf(s_gates[2 * EPW + lane]);
        float og = sigm(s_gates[3 * EPW + lane]);
        c2 = fg * c2 + ig * gg;
        g_h2[e0 + lane] = og * tanhf(c2);
    }
    bar += NWG; grid_barrier(cnt, bar);
    for (int i = tid; i < H; i += TPB) s_h2[i] = g_h2[i];
    for (int i = tid; i < DW; i += TPB) s_word[i] = emb[i];   // BOS = 0
    __syncthreads();

    // ---- decode loop
    for (int step = 0; step < STEPS; ++step) {
        // LSTM1: x = pad_vid = 0  -> gates = h1 @ w_hh1^T + biases
        for (int l = wave; l < LROWS; l += 16) {
            const int r = (l >> 5) * H + e0 + (l & 31);
            const float* wr = w_hh1 + (size_t)r * H;
            float acc = 0.f;
            for (int k = lane * 4; k < H; k += 128) {
                float4 w4 = *(const float4*)(wr + k);
                acc += w4.x * s_h1[k] + w4.y * s_h1[k + 1] + w4.z * s_h1[k + 2] + w4.w * s_h1[k + 3];
            }
            for (int off = 16; off; off >>= 1) acc += __shfl_xor(acc, off, 32);
            if (lane == 0) s_gates[l] = acc + b_ih1[r] + b_hh1[r];
        }
        __syncthreads();
        if (wave == 0) {
            float ig = sigm(s_gates[lane]);
            float fg = sigm(s_gates[EPW + lane]);
            float gg = tanhf(s_gates[2 * EPW + lane]);
            float og = sigm(s_gates[3 * EPW + lane]);
            c1 = fg * c1 + ig * gg;
            g_h1[e0 + lane] = og * tanhf(c1);
        }
        bar += NWG; grid_barrier(cnt, bar);                 // barrier #1
        for (int i = tid; i < H; i += TPB) s_h1[i] = g_h1[i];
        __syncthreads();

        // LSTM2: gates = [h1, word] @ w_ih2^T + h2 @ w_hh2^T + biases
        for (int l = wave; l < LROWS; l += 16) {
            const int r = (l >> 5) * H + e0 + (l & 31);
            const float* wi = w_ih2 + (size_t)r * (H + DW);
            const float* wh = w_hh2 + (size_t)r * H;
            float acc = 0.f;
            for (int k = lane * 4; k < H; k += 128) {
                float4 a4 = *(const float4*)(wi + k);
                acc += a4.x * s_h1[k] + a4.y * s_h1[k + 1] + a4.z * s_h1[k + 2] + a4.w * s_h1[k + 3];
                float4 b4 = *(const float4*)(wh + k);
                acc += b4.x * s_h2[k] + b4.y * s_h2[k + 1] + b4.z * s_h2[k + 2] + b4.w * s_h2[k + 3];
            }
            for (int k = lane * 4; k < DW; k += 128) {
                float4 w4 = *(const float4*)(wi + H + k);
                acc += w4.x * s_word[k] + w4.y * s_word[k + 1] + w4.z * s_word[k + 2] + w4.w * s_word[k + 3];
            }
            for (int off = 16; off; off >>= 1) acc += __shfl_xor(acc, off, 32);
            if (lane == 0) s_gates[l] = acc + b_ih2[r] + b_hh2[r];
        }
        __syncthreads();
        if (wave == 0) {
            float ig = sigm(s_gates[lane]);
            float fg = sigm(s_gates[EPW + lane]);
            float gg = tanhf(s_gates[2 * EPW + lane]);
            float og = sigm(s_gates[3 * EPW + lane]);
            c2 = fg * c2 + ig * gg;
            float hv = og * tanhf(c2);
            const int j = e0 + lane;
            g_h2[j] = hv;
            H2out[(size_t)step * H + j] = hv;
            // local argmax over this WG's 32 elements (ties -> lowest index)
            float bv = hv; int bi = j;
            for (int off = 16; off; off >>= 1) {
                float ov = __shfl_xor(bv, off, 32);
                int   oi = __shfl_xor(bi, off, 32);
                if (ov > bv || (ov == bv && oi < bi)) { bv = ov; bi = oi; }
            }
            if (lane == 0) { g_redv[wg] = bv; g_redi[wg] = bi; }
        }
        bar += NWG; grid_barrier(cnt, bar);                 // barrier #2
        for (int i = tid; i < H; i += TPB) s_h2[i] = g_h2[i];
        if (wave == 0) {
            // every WG reduces the same 32 partials identically -> same idx
            float v = g_redv[lane]; int i = g_redi[lane];
            for (int off = 16; off; off >>= 1) {
                float ov = __shfl_xor(v, off, 32);
                int   oi = __shfl_xor(i, off, 32);
                if (ov > v || (ov == v && oi < i)) { v = ov; i = oi; }
            }
            if (lane == 0) s_idx = i;
        }
        __syncthreads();
        const int idx = s_idx;
        for (int i = tid; i < DW; i += TPB) s_word[i] = emb[(size_t)idx * DW + i];
        __syncthreads();
    }
}

// ---------------------------------------------------------------------------
// Kernel 2: logits = H2pad(48x1024) @ w_out^T (+b_out) via V_WMMA_F32_16X16X4_F32.
// Each block owns a 64-wide N strip; 12 waves = 3 M-tiles x 4 N-tiles.
// w_out is streamed from HBM exactly once (131 MB total).
// ---------------------------------------------------------------------------
__global__ __launch_bounds__(384) void logits_wmma_kernel(
    const float* __restrict__ H2, const float* __restrict__ w_out,
    const float* __restrict__ b_out, float* __restrict__ out)
{
    __shared__ float Ablk[MPAD][132];   // +4 pad: conflict-free column reads
    __shared__ float Bblk[64][132];

    const int tid  = threadIdx.x;
    const int lane = tid & 31;
    const int wave = tid >> 5;          // 0..11
    const int mi   = wave % 3;          // M tile
    const int ni   = wave / 3;          // N tile
    const int n0   = blockIdx.x * 64;
    const int half = lane >> 4;         // lanes 16-31 hold K+2/K+3
    const int l16  = lane & 15;

    v8f acc = {0.f, 0.f, 0.f, 0.f, 0.f, 0.f, 0.f, 0.f};

    for (int ko = 0; ko < 1024; ko += 128) {
        // stage A: 48 x 128
        for (int i = tid; i < MPAD * 32; i += 384) {
            int m = i >> 5, k4 = (i & 31) << 2;
            *(float4*)&Ablk[m][k4] = *(const float4*)(H2 + (size_t)m * 1024 + ko + k4);
        }
        // stage B: 64 rows of w_out x 128 k (B[k][n] = w_out[n][k])
        for (int i = tid; i < 64 * 32; i += 384) {
            int n = i >> 5, k4 = (i & 31) << 2;
            *(float4*)&Bblk[n][k4] = *(const float4*)(w_out + (size_t)(n0 + n) * 1024 + ko + k4);
        }
        if (ko + 128 < 1024)
            __builtin_prefetch(w_out + (size_t)(n0 + (tid >> 3)) * 1024 + ko + 128, 0, 1);
        __syncthreads();

        const float* arow = &Ablk[mi * 16 + l16][0];
        const float* brow = &Bblk[ni * 16 + l16][0];
        #pragma unroll 8
        for (int kk = 0; kk < 128; kk += 4) {
            v2f a, b;
            a.x = arow[kk + 2 * half];
            a.y = arow[kk + 2 * half + 1];
            b.x = brow[kk + 2 * half];
            b.y = brow[kk + 2 * half + 1];
            acc = __builtin_amdgcn_wmma_f32_16x16x4_f32(
                /*neg_a=*/false, a, /*neg_b=*/false, b,
                /*c_mod=*/(short)0, acc, /*reuse_a=*/false, /*reuse_b=*/false);
        }
        __syncthreads();
    }

    const int n    = n0 + ni * 16 + l16;
    const float bias = b_out[n];
    #pragma unroll
    for (int r = 0; r < 8; ++r) {
        int m = mi * 16 + r + 8 * half;   // C/D layout: vgpr r -> M=r (lanes<16) / M=r+8
        if (m < STEPS) out[(size_t)m * VOCAB + n] = acc[r] + bias;
    }
}

// ---------------------------------------------------------------------------
// Kernel 3: in-place log_softmax over each 32000-wide row of d_out.
// ---------------------------------------------------------------------------
__global__ __launch_bounds__(512) void logsoftmax_kernel(float* __restrict__ out)
{
    __shared__ float red[16];
    const int row  = blockIdx.x;
    float* p = out + (size_t)row * VOCAB;
    const int tid = threadIdx.x, lane = tid & 31, wave = tid >> 5;

    float m = -__builtin_inff();
    for (int i = tid; i < VOCAB; i += 512) m = fmaxf(m, p[i]);
    for (int off = 16; off; off >>= 1) m = fmaxf(m, __shfl_xor(m, off, 32));
    if (lane == 0) red[wave] = m;
    __syncthreads();
    float rm = red[lane & 15];
    for (int off = 8; off; off >>= 1) rm = fmaxf(rm, __shfl_xor(rm, off, 32));
    __syncthreads();   // everyone done reading red before reuse

    float s = 0.f;
    for (int i = tid; i < VOCAB; i += 512) s += expf(p[i] - rm);
    for (int off = 16; off; off >>= 1) s += __shfl_xor(s, off, 32);
    if (lane == 0) red[wave] = s;
    __syncthreads();
    float rs = red[lane & 15];
    for (int off = 8; off; off >>= 1) rs += __shfl_xor(rs, off, 32);

    const float lg = logf(rs) + rm;
    for (int i = tid; i < VOCAB; i += 512) p[i] = p[i] - lg;
}

// ---------------------------------------------------------------------------
extern "C" void kernel_launch(void* const* d_in, const int* in_sizes, int n_in,
                              void* d_out, int out_size, void* d_ws, size_t ws_size,
                              hipStream_t stream)
{
    const float* vid   = (const float*)d_in[0];
    const float* w_ih1 = (const float*)d_in[1];
    const float* w_hh1 = (const float*)d_in[2];
    const float* b_ih1 = (const float*)d_in[3];
    const float* b_hh1 = (const float*)d_in[4];
    const float* w_ih2 = (const float*)d_in[5];
    const float* w_hh2 = (const float*)d_in[6];
    const float* b_ih2 = (const float*)d_in[7];
    const float* b_hh2 = (const float*)d_in[8];
    const float* emb   = (const float*)d_in[9];
    const float* w_out = (const float*)d_in[10];
    const float* b_out = (const float*)d_in[11];

    float* out = (float*)d_out;

    // workspace layout (floats)
    float*    F      = (float*)d_ws;
    float*    H2     = F;                       // MPAD*H = 49152 floats
    float*    g_h1   = F + (size_t)MPAD * H;    // 1024
    float*    g_h2   = g_h1 + H;                // 1024
    float*    g_redv = g_h2 + H;                // NWG floats
    int*      g_redi = (int*)(g_redv + NWG);    // NWG ints
    unsigned* cnt    = (unsigned*)(g_redi + NWG);

    hipLaunchKernelGGL(init_kernel, dim3(1), dim3(256), 0, stream, H2, cnt);

    hipLaunchKernelGGL(lstm_seq_kernel, dim3(NWG), dim3(TPB), 0, stream,
                       vid, w_ih1, w_hh1, b_ih1, b_hh1,
                       w_ih2, w_hh2, b_ih2, b_hh2, emb,
                       H2, g_h1, g_h2, g_redv, g_redi, cnt);

    hipLaunchKernelGGL(logits_wmma_kernel, dim3(VOCAB / 64), dim3(384), 0, stream,
                       H2, w_out, b_out, out);

    hipLaunchKernelGGL(logsoftmax_kernel, dim3(STEPS), dim3(512), 0, stream, out);
}